// CostVolume_72602127171752
// MI455X (gfx1250) — compile-verified
//
#include <hip/hip_runtime.h>

typedef __attribute__((ext_vector_type(2))) float v2f;
typedef __attribute__((ext_vector_type(8))) float v8f;

namespace {
constexpr int W      = 240;   // width
constexpr int H      = 128;   // height
constexpr int C      = 128;   // channels (K of the GEMM)
constexpr int V      = 48;    // disparity volume
constexpr int CHUNK  = 32;    // channels per K chunk staged in LDS
constexpr int NCHUNK = C / CHUNK;
constexpr int SLABP  = 241;   // padded slab row (gcd(15,64)=1 -> conflict-free scatter)
constexpr int NWAVE  = 15;    // one wave per 16-wide M tile (15*16 = 240)
constexpr int NTHR   = NWAVE * 32;
constexpr int ROWV4  = W / 4;        // 60 float4 per channel row
constexpr int CSTRIDE4 = H * ROWV4;  // float4 stride between channels
}

__global__ __launch_bounds__(NTHR)
void cost_volume_wmma_kernel(const float* __restrict__ L,
                             const float* __restrict__ R,
                             float* __restrict__ out) {
  __shared__ float sL[CHUNK * W];
  __shared__ float sR[CHUNK * W];
  __shared__ float slab[V * SLABP];

  const int tid  = threadIdx.x;
  const int lane = tid & 31;
  const int hi   = lane >> 4;   // 0/1 : selects K pair {0,1} vs {2,3}
  const int lo   = lane & 15;   // M (A) or N (B) index within tile
  const int wave = tid >> 5;    // 0..14
  const int x0   = wave * 16;   // M-tile origin in x

  const int b = blockIdx.x / H;
  const int h = blockIdx.x % H;

  // zero the output slab (masked x<i positions stay zero)
  for (int idx = tid; idx < V * SLABP; idx += NTHR) slab[idx] = 0.0f;

  const float4* L4 = reinterpret_cast<const float4*>(L);
  const float4* R4 = reinterpret_cast<const float4*>(R);
  float4* sL4 = reinterpret_cast<float4*>(sL);
  float4* sR4 = reinterpret_cast<float4*>(sR);

  // base float4 index of (b, c=0, h, x=0)
  const size_t rowBase = ((size_t)b * C * H + h) * ROWV4;

  v8f acc[4] = {};  // 4 diagonal-offset tiles per wave

  // per-thread staging assignment: 4 float4 per array per chunk
  int lrow[4], lcol[4];
#pragma unroll
  for (int r = 0; r < 4; ++r) {
    int idx = tid + NTHR * r;       // 0..1919
    lrow[r] = idx / ROWV4;          // channel within chunk (0..31)
    lcol[r] = idx % ROWV4;          // float4 column (0..59)
  }

  float4 bufL[4], bufR[4];
#pragma unroll
  for (int r = 0; r < 4; ++r) {     // prefetch chunk 0
    size_t off = rowBase + (size_t)lrow[r] * CSTRIDE4 + lcol[r];
    bufL[r] = L4[off];
    bufR[r] = R4[off];
  }

#pragma unroll 1
  for (int ch = 0; ch < NCHUNK; ++ch) {
    __syncthreads();                 // prior compute / slab-zero complete
#pragma unroll
    for (int r = 0; r < 4; ++r) {
      sL4[lrow[r] * ROWV4 + lcol[r]] = bufL[r];
      sR4[lrow[r] * ROWV4 + lcol[r]] = bufR[r];
    }
    __syncthreads();

    if (ch + 1 < NCHUNK) {           // issue next chunk's global loads early
      const size_t cb = rowBase + (size_t)(ch + 1) * CHUNK * CSTRIDE4;
#pragma unroll
      for (int r = 0; r < 4; ++r) {
        size_t off = cb + (size_t)lrow[r] * CSTRIDE4 + lcol[r];
        bufL[r] = L4[off];
        bufR[r] = R4[off];
      }
    }

    // 8 K-steps of 4 over this chunk
#pragma unroll
    for (int k = 0; k < CHUNK; k += 4) {
      v2f a;                                   // A = L^T fragment (16x4)
      a.x = sL[(k + 2 * hi    ) * W + x0 + lo];
      a.y = sL[(k + 2 * hi + 1) * W + x0 + lo];
#pragma unroll
      for (int d = 0; d < 4; ++d) {
        int xp0 = x0 - 16 * d;
        if (xp0 < 0) xp0 = 0;                  // clamp: garbage tile, discarded later
        v2f bv;                                // B = R fragment (4x16)
        bv.x = sR[(k + 2 * hi    ) * W + xp0 + lo];
        bv.y = sR[(k + 2 * hi + 1) * W + xp0 + lo];
        acc[d] = __builtin_amdgcn_wmma_f32_16x16x4_f32(
            false, a, false, bv, (short)0, acc[d], false, false);
      }
    }
  }

  // scatter accumulators into the banded slab: i = m - n + 16*d, x = x0 + m
#pragma unroll
  for (int d = 0; d < 4; ++d) {
    if (x0 >= 16 * d) {                        // tile fully inside valid x' range
#pragma unroll
      for (int r = 0; r < 8; ++r) {
        int m = r + 8 * hi;
        int i = m - lo + 16 * d;
        if (i >= 0 && i < V) slab[i * SLABP + x0 + m] = acc[d][r];
      }
    }
  }

  __syncthreads();

  // coalesced store with mean scale
  const float scale = 1.0f / (float)C;
  const size_t outBase = ((size_t)b * V * H + h) * W;
  for (int idx = tid; idx < V * W; idx += NTHR) {
    int i = idx / W;
    int x = idx - i * W;
    out[outBase + (size_t)i * (H * W) + x] = slab[i * SLABP + x] * scale;
  }
}

extern "C" void kernel_launch(void* const* d_in, const int* in_sizes, int n_in,
                              void* d_out, int out_size, void* d_ws, size_t ws_size,
                              hipStream_t stream) {
  const float* L = (const float*)d_in[0];
  const float* R = (const float*)d_in[1];
  float* out = (float*)d_out;
  (void)in_sizes; (void)n_in; (void)out_size; (void)d_ws; (void)ws_size;

  dim3 grid(8 * 128);   // one workgroup per (b, h)
  dim3 block(NTHR);     // 15 waves of 32
  hipLaunchKernelGGL(cost_volume_wmma_kernel, grid, block, 0, stream, L, R, out);
}